// QLSTM_65481071399364
// MI455X (gfx1250) — compile-verified
//
#include <hip/hip_runtime.h>
#include <stdint.h>

#define NQ     8
#define DIN    128
#define DK     136           // D + H
#define NOUT   32            // 4 gates * 8 qubits stacked
#define TSTEPS 64
#define BATCH  1024
#define RPW    16            // batch rows per wave (one WMMA M-tile)
#define WPB    2             // waves per block
#define RPB    (RPW*WPB)

typedef float v2f __attribute__((ext_vector_type(2)));
typedef float v8f __attribute__((ext_vector_type(8)));

__device__ __forceinline__ float sigm(float x){ return 1.0f/(1.0f + __expf(-x)); }

// Issue one 16-row x-tile (16 x 128 floats) as 16 async B128 transfers:
// each lane moves 16 bytes global -> LDS directly (no VGPR staging), ASYNCcnt += 16.
__device__ __forceinline__ void issue_xtile_async(const float* __restrict__ xin,
                                                  float* ldsComb, int t, int r0, int lane)
{
    const uint64_t gbase = (uint64_t)(uintptr_t)(xin + ((size_t)t*BATCH + r0)*DIN)
                         + (uint64_t)(lane * 16);
    const uint32_t lbase = (uint32_t)(uintptr_t)ldsComb + (uint32_t)(lane * 16);
    #pragma unroll
    for (int i = 0; i < RPW; ++i) {
        uint32_t laddr = lbase + (uint32_t)(i * DK * 4);     // row stride 544 B in LDS
        uint64_t gaddr = gbase + (uint64_t)(i * DIN * 4);    // row stride 512 B in HBM
        asm volatile("global_load_async_to_lds_b128 %0, %1, off"
                     :: "v"(laddr), "v"(gaddr) : "memory");
    }
}

__global__ __launch_bounds__(WPB*32, 1)
void qlstm_kernel(const float* __restrict__ xin,
                  const float* __restrict__ Wf, const float* __restrict__ bf, const float* __restrict__ Pf,
                  const float* __restrict__ Wi, const float* __restrict__ bi, const float* __restrict__ Pi,
                  const float* __restrict__ Wu, const float* __restrict__ bu, const float* __restrict__ Pu,
                  const float* __restrict__ Wo, const float* __restrict__ bo, const float* __restrict__ Po,
                  float* __restrict__ out)
{
    __shared__ float sW[NOUT*DK];           // stacked W (gate-major), B-matrix source
    __shared__ float sB[NOUT];
    __shared__ float sCP[NOUT];             // cos(P[:,0])
    __shared__ float sComb[WPB][2][RPW*DK]; // double-buffered per-wave [x_t | h] tiles

    const int tid  = threadIdx.x;
    const int lane = tid & 31;
    const int wv   = tid >> 5;
    const int lo16 = lane & 15;
    const int hiK  = lane >> 4;          // 0/1: K-pair for A/B, row-half for C
    const int q8   = lane & 7;           // qubit index within gate (column group)
    const bool hi8 = (lane & 8) != 0;
    const int r0   = blockIdx.x * RPB + wv * RPW;
    float* buf0 = &sComb[wv][0][0];
    float* buf1 = &sComb[wv][1][0];

    // kick off step-0 x tile immediately (overlaps with W staging below)
    issue_xtile_async(xin, buf0, 0, r0, lane);

    // ---- stage weights / bias / cos(P) into LDS ----
    {
        const float* Wp[4] = {Wf, Wi, Wu, Wo};
        for (int idx = tid; idx < NOUT*DK; idx += WPB*32) {
            int n = idx / DK, k = idx - n*DK;
            sW[idx] = Wp[n>>3][(n&7)*DK + k];
        }
        if (tid < NOUT) {
            int g = tid >> 3, q = tid & 7;
            const float* bp = (g==0)?bf:(g==1)?bi:(g==2)?bu:bo;
            const float* Pp = (g==0)?Pf:(g==1)?Pi:(g==2)?Pu:Po;
            sB[tid]  = bp[q];
            sCP[tid] = __cosf(Pp[q*3]);  // only P[:,0] survives the algebra
        }
    }
    // zero h columns of the step-0 tile (h0 = 0); disjoint from async x columns
    for (int idx = lane; idx < RPW*NQ; idx += 32)
        buf0[(idx>>3)*DK + DIN + (idx&7)] = 0.0f;
    __syncthreads();

    const float bias0 = sB[lo16],  bias1 = sB[16+lo16];
    const float cp0   = sCP[lo16], cp1   = sCP[16+lo16];

    float cst[8], hreg[8];
    #pragma unroll
    for (int j = 0; j < 8; ++j) { cst[j] = 0.0f; hreg[j] = 0.0f; }

    for (int t = 0; t < TSTEPS; ++t) {
        float* cur = (t & 1) ? buf1 : buf0;
        float* nxt = (t & 1) ? buf0 : buf1;

        // prefetch next x tile; then wait for current tile (16 newest ops may remain in flight)
        if (t + 1 < TSTEPS) {
            issue_xtile_async(xin, nxt, t + 1, r0, lane);
            asm volatile("s_wait_asynccnt 0x10" ::: "memory");
        } else {
            asm volatile("s_wait_asynccnt 0x0" ::: "memory");
        }

        // ---- theta = comb @ W^T + b via V_WMMA_F32_16X16X4_F32 (K=136 = 34 x 4) ----
        v8f c0, c1;
        #pragma unroll
        for (int j = 0; j < 8; ++j) { c0[j] = bias0; c1[j] = bias1; }
        const int ak = 2*hiK;
        #pragma unroll
        for (int kk = 0; kk < DK; kk += 4) {
            v2f a  = *(const v2f*)&cur[lo16*DK       + kk + ak]; // A: 16x4 rows of comb
            v2f b0 = *(const v2f*)&sW [lo16*DK       + kk + ak]; // B: 4x16, n = 0..15
            v2f b1 = *(const v2f*)&sW [(16+lo16)*DK  + kk + ak]; // B: 4x16, n = 16..31
            c0 = __builtin_amdgcn_wmma_f32_16x16x4_f32(false, a, false, b0, (short)0, c0, false, false);
            c1 = __builtin_amdgcn_wmma_f32_16x16x4_f32(false, a, false, b1, (short)0, c1, false, false);
        }

        // ---- qgate collapsed: z = cos(P)cos(theta); segmented cumprod over qubits ----
        v8f z0, z1;
        #pragma unroll
        for (int j = 0; j < 8; ++j) { z0[j] = cp0*__cosf(c0[j]); z1[j] = cp1*__cosf(c1[j]); }
        #pragma unroll
        for (int d = 1; d < 8; d <<= 1) {
            #pragma unroll
            for (int j = 0; j < 8; ++j) {
                float u0 = __shfl_up(z0[j], (unsigned)d, 8);
                float u1 = __shfl_up(z1[j], (unsigned)d, 8);
                if (q8 >= d) { z0[j] *= u0; z1[j] *= u1; }
            }
        }

        // ---- gates: tile0 cols 0-7 = f, 8-15 = i ; tile1 cols 0-7 = u, 8-15 = o ----
        #pragma unroll
        for (int j = 0; j < 8; ++j) {
            float t0 = z0[j], t1 = z1[j];
            float p0 = __shfl_xor(t0, 8, 32);
            float p1 = __shfl_xor(t1, 8, 32);
            float fg = hi8 ? p0 : t0;
            float ig = hi8 ? t0 : p0;
            float gg = hi8 ? p1 : t1;
            float og = hi8 ? t1 : p1;
            fg = sigm(fg); ig = sigm(ig); gg = tanhf(gg); og = sigm(og);
            float cn = fg*cst[j] + ig*gg;
            cst[j] = cn;
            float hn = og*tanhf(cn);
            hreg[j] = hn;
            if (!hi8) {                               // one writer per (row,qubit)
                int row = j + 8*hiK;
                nxt[row*DK + DIN + q8] = hn;          // h feeds next step's A-matrix
                out[((size_t)t*BATCH + (r0+row))*NQ + q8] = hn;
            }
        }
    }

    // ---- final hx, cx appended after outputs ----
    float* hx = out + (size_t)TSTEPS*BATCH*NQ;
    float* cx = hx + (size_t)BATCH*NQ;
    if (!hi8) {
        #pragma unroll
        for (int j = 0; j < 8; ++j) {
            int row = j + 8*hiK;
            hx[(size_t)(r0+row)*NQ + q8] = hreg[j];
            cx[(size_t)(r0+row)*NQ + q8] = cst[j];
        }
    }
}

extern "C" void kernel_launch(void* const* d_in, const int* in_sizes, int n_in,
                              void* d_out, int out_size, void* d_ws, size_t ws_size,
                              hipStream_t stream) {
    (void)in_sizes; (void)n_in; (void)out_size; (void)d_ws; (void)ws_size;
    const float* xin = (const float*)d_in[0];
    const float* Wf = (const float*)d_in[1];  const float* bf = (const float*)d_in[2];  const float* Pf = (const float*)d_in[3];
    const float* Wi = (const float*)d_in[4];  const float* bi = (const float*)d_in[5];  const float* Pi = (const float*)d_in[6];
    const float* Wu = (const float*)d_in[7];  const float* bu = (const float*)d_in[8];  const float* Pu = (const float*)d_in[9];
    const float* Wo = (const float*)d_in[10]; const float* bo = (const float*)d_in[11]; const float* Po = (const float*)d_in[12];
    float* out = (float*)d_out;
    dim3 grid(BATCH / RPB), block(WPB * 32);
    hipLaunchKernelGGL(qlstm_kernel, grid, block, 0, stream,
                       xin, Wf, bf, Pf, Wi, bi, Pi, Wu, bu, Pu, Wo, bo, Po, out);
}